// RRT_14998025798477
// MI455X (gfx1250) — compile-verified
//
#include <hip/hip_runtime.h>
#include <stdint.h>

// RRT tree growth: 8192-iteration serial chain, each step = exact argmin over
// the current tree + one steered node append. Single workgroup on one WGP,
// tree + samples fully LDS-resident (131 KB of the 320 KB CDNA5 WGP LDS).
// One workgroup barrier per iteration; argmin keys reduced with v_min_num_f64
// (u64 key order == positive-double order) + one ds_min_u64 per wave.

#define MAX_ITER 8192
#define NTHREADS 256   // 8 wave32 waves on one WGP (4 SIMD32s)

// Dynamic LDS layout:
//   [0,      65536)  float2 nodes[8192]   (indices 0..8191; node 8192 never scanned)
//   [65536, 131072)  float2 samples[8192] (staged via async-to-LDS)
//   [131072,131096)  u64 cells[3]         (period-3 rotating argmin cells)
#define NODES_OFF    0
#define SAMPLES_OFF  65536
#define CELL_OFF     131072
#define SMEM_BYTES   131096

// Sentinel/init key: hi=0x7F800000 (f32 +inf bits -> large positive finite
// double), lo=0xFFFFFFFF. Larger than every real key (hi < 0x48800000).
#define KEY_INIT 0x7F800000FFFFFFFFull

// ---------------------------------------------------------------------------
// Kernel 1: samples[i] = u[i]<0.1 ? goal : r[i]*200 (bit-exact: lo=0, hi-lo=200
// -> lo + r*(hi-lo) == r*200, unfused). Also writes the root node to out.
// ---------------------------------------------------------------------------
__global__ __launch_bounds__(NTHREADS) void rrt_prep(
    const float* __restrict__ state, const float* __restrict__ goal,
    const float* __restrict__ u, const float* __restrict__ r,
    float* __restrict__ samples, float* __restrict__ out)
{
  int i = blockIdx.x * blockDim.x + threadIdx.x;
  if (i < MAX_ITER) {
    const bool g = u[i] < 0.1f;
    const float sx = g ? goal[0] : __fmul_rn(r[2 * i],     200.0f);
    const float sy = g ? goal[1] : __fmul_rn(r[2 * i + 1], 200.0f);
    samples[2 * i]     = sx;
    samples[2 * i + 1] = sy;
  }
  if (i == 0) { out[0] = state[0]; out[1] = state[1]; }
}

// ---------------------------------------------------------------------------
// Kernel 2: persistent single-workgroup sequential tree build.
// ---------------------------------------------------------------------------
__global__ __launch_bounds__(NTHREADS) void rrt_main(
    const float* __restrict__ state,
    const float* __restrict__ samples_g,
    float* __restrict__ out)
{
  extern __shared__ char smem[];
  float2* nodes   = (float2*)(smem + NODES_OFF);
  float2* samples = (float2*)(smem + SAMPLES_OFF);
  unsigned long long* cells = (unsigned long long*)(smem + CELL_OFF);

  const int tid = threadIdx.x;

  // ---- Stage all 8192 samples (64 KB) global -> LDS with CDNA5 async copies.
  // 256 lanes x 16 B x 16 issues = 64 KB; each wave waits its own ASYNCcnt,
  // the workgroup barrier below joins them.
  {
    const uint32_t lds_base = (uint32_t)(uintptr_t)(smem + SAMPLES_OFF);
    const uint64_t g_base   = (uint64_t)(uintptr_t)samples_g;
#pragma unroll
    for (int k = 0; k < 16; ++k) {
      const uint32_t off = (uint32_t)((k * NTHREADS + tid) * 16);
      const uint32_t l = lds_base + off;
      const uint64_t g = g_base + off;
      asm volatile("global_load_async_to_lds_b128 %0, %1, off"
                   :: "v"(l), "v"(g) : "memory");
    }
    asm volatile("s_wait_asynccnt 0x0" ::: "memory");
  }

  // Sentinel-fill the tree so scans can run to a uniform padded bound:
  // +inf coords -> d2 = +inf -> key always loses against any real node.
  const float INF = __int_as_float(0x7F800000);
  for (int j = tid; j < MAX_ITER; j += NTHREADS) nodes[j] = make_float2(INF, INF);

  const float2 start = make_float2(state[0], state[1]);
  if (tid == 0) {
    nodes[0] = start;                        // same thread as the fill: in-order
    cells[0] = KEY_INIT; cells[1] = KEY_INIT; cells[2] = KEY_INIT;
  }
  __syncthreads();

  float px = start.x, py = start.y;          // register copy of newest node
  int c = 0;                                 // rotating cell index (mod 3)

  for (int i = 0; i < MAX_ITER; ++i) {
    const float2 s = samples[i];             // LDS broadcast read
    const int K = (i >> 8) + 1;              // uniform scalar trip count

    // Local argmin over strided share of nodes[0 .. K*256). Slot i may race
    // with its own store (benign: duplicate exact key or inf sentinel);
    // slots > i are sentinels. Key = positive double with bits
    // (f32bits(d2)<<32)|j: fmin == u64-min == (min d2, lowest idx on ties),
    // matching jnp.argmin bit-exactly. d2 unfused to mirror XLA's reduce.
    double best = __longlong_as_double((long long)KEY_INIT);
    int j = tid;
    for (int k = 0; k < K; ++k) {
      const float2 n = nodes[j];                       // ds_load_b64, bank-clean
      const float dx = __fsub_rn(n.x, s.x);
      const float dy = __fsub_rn(n.y, s.y);
      const float d2 = __fadd_rn(__fmul_rn(dx, dx), __fmul_rn(dy, dy));
      best = fmin(best, __hiloint2double((int)__float_as_uint(d2), j));
      j += NTHREADS;
    }
    // Exactly one thread evaluates the newest node (index i) from registers,
    // covering the case where its LDS store isn't visible yet.
    if (tid == (i & (NTHREADS - 1))) {
      const float dx = __fsub_rn(px, s.x);
      const float dy = __fsub_rn(py, s.y);
      const float d2 = __fadd_rn(__fmul_rn(dx, dx), __fmul_rn(dy, dy));
      best = fmin(best, __hiloint2double((int)__float_as_uint(d2), i));
    }

    // wave32 butterfly (v_min_num_f64), then one ds_min_u64 per wave.
#pragma unroll
    for (int o = 16; o > 0; o >>= 1)
      best = fmin(best, __shfl_xor(best, o, 32));
    if ((tid & 31) == 0)
      atomicMin(&cells[c], (unsigned long long)__double_as_longlong(best));

    __syncthreads();   // the ONLY barrier per iteration

    // Everyone redundantly computes the steered node (keeps prev in registers).
    const unsigned long long key = ((volatile unsigned long long*)cells)[c];
    const int nn = (int)(unsigned)key;
    const float d2 = __uint_as_float((unsigned)(key >> 32)); // (a-b)^2==(b-a)^2
    const float2 from = nodes[nn];           // visible: stored >=1 barrier ago
    const float dirx = __fsub_rn(s.x, from.x);
    const float diry = __fsub_rn(s.y, from.y);
    const float dist = sqrtf(__fadd_rn(d2, 1e-12f));
    const float scale = dist > 5.0f ? __fdiv_rn(5.0f, dist) : 1.0f;
    px = __fadd_rn(from.x, __fmul_rn(dirx, scale));
    py = __fadd_rn(from.y, __fmul_rn(diry, scale));

    if (tid == 0) {
      if (i + 1 < MAX_ITER) nodes[i + 1] = make_float2(px, py);
      out[2 * (i + 1)]     = px;
      out[2 * (i + 1) + 1] = py;
      // Reset the cell used LAST iteration: its reads are fenced by the
      // barrier above; its next atomics are fenced by the next barrier.
      cells[c == 0 ? 2 : c - 1] = KEY_INIT;
    }
    c = (c == 2) ? 0 : c + 1;
  }
}

// ---------------------------------------------------------------------------
extern "C" void kernel_launch(void* const* d_in, const int* in_sizes, int n_in,
                              void* d_out, int out_size, void* d_ws, size_t ws_size,
                              hipStream_t stream) {
  (void)in_sizes; (void)n_in; (void)out_size; (void)ws_size;
  const float* state = (const float*)d_in[0];   // (3,)
  const float* goal  = (const float*)d_in[1];   // (2,)
  const float* u     = (const float*)d_in[2];   // (8192,)
  const float* r     = (const float*)d_in[3];   // (8192,2)
  float* out     = (float*)d_out;               // (8193,2)
  float* samples = (float*)d_ws;                // 8192 float2 = 64 KB scratch

  (void)hipFuncSetAttribute((const void*)rrt_main,
                            hipFuncAttributeMaxDynamicSharedMemorySize,
                            SMEM_BYTES);

  rrt_prep<<<(MAX_ITER + NTHREADS - 1) / NTHREADS, NTHREADS, 0, stream>>>(
      state, goal, u, r, samples, out);
  rrt_main<<<1, NTHREADS, SMEM_BYTES, stream>>>(state, samples, out);
}